// CliffordAlgebra_10290741641283
// MI455X (gfx1250) — compile-verified
//
#include <hip/hip_runtime.h>

typedef __attribute__((ext_vector_type(2))) float v2f;
typedef __attribute__((ext_vector_type(4))) float v4f;
typedef __attribute__((ext_vector_type(8))) float v8f;

// Geometric product out[n,k] = sum_{i,j} a[n,i] b[n,j] C[i,j,k]  (8 blades)
// GEMM form: P[n, f=i*8+j] = a[n,i]*b[n,j];  W[f,k] = C[i,j,k];  out = P @ W
// via chained V_WMMA_F32_16X16X4_F32 (K = 64 -> 16 chunks of 4).
// Main kernel handles full 16-mv tiles only (no per-store guards); a scalar
// tail kernel covers any remainder (none for the harness batch size).
__global__ __launch_bounds__(256) void clifford_gp_wmma(
    const float* __restrict__ a,
    const float* __restrict__ b,
    const float* __restrict__ cayley,   // [8][8][8] row-major, k fastest
    float* __restrict__ out,
    long long n_full_tiles)
{
    const int lane = threadIdx.x & 31;
    const int wave = threadIdx.x >> 5;
    const int half = lane >> 4;     // 0: lanes 0-15, 1: lanes 16-31
    const int n    = lane & 15;     // tile row (for loads) / D column
    const int o    = half * 2;      // this half's K sub-offset within a chunk

    // ---- Constant B fragments (W matrix), built once per wave ----
    // Chunk c covers K rows f = 4c..4c+3; this lane supplies rows f0 = 4c+o
    // (.x) and f0+1 (.y), column n. Columns 8..15 are zero padding.
    v2f bfrag[16];
    #pragma unroll
    for (int c = 0; c < 16; ++c) {
        float w0 = 0.0f, w1 = 0.0f;
        if (n < 8) {
            const int f0 = 4 * c + o;
            const int f1 = f0 + 1;
            w0 = cayley[(f0 >> 3) * 64 + (f0 & 7) * 8 + n];
            w1 = cayley[(f1 >> 3) * 64 + (f1 & 7) * 8 + n];
        }
        bfrag[c].x = w0;
        bfrag[c].y = w1;
    }

    const long long stride = (long long)gridDim.x * 8;

    for (long long t = (long long)blockIdx.x * 8 + wave; t < n_full_tiles;
         t += stride) {
        // Lane loads multivector m = t*16 + n (lanes n and n+16 share one mv:
        // a duplicated via cache, b split: this lane needs j in {o,o+1,o+4,o+5}).
        const long long m  = t * 16 + n;
        const float* ap = a + m * 8;
        const float* bp = b + m * 8;

        const v4f a0  = *(const v4f*)(ap);
        const v4f a1  = *(const v4f*)(ap + 4);
        const v2f bl0 = *(const v2f*)(bp + o);      // b[o],   b[o+1]
        const v2f bl1 = *(const v2f*)(bp + o + 4);  // b[o+4], b[o+5]
        const float areg[8] = {a0.x, a0.y, a0.z, a0.w, a1.x, a1.y, a1.z, a1.w};

        v8f acc = {};   // 16x16 f32 accumulator (cols 8..15 unused)

        // For chunk c this lane's A elements are
        //   P[m, 4c+o]   = a[m, c>>1] * b[m, (c&1)*4 + o]
        //   P[m, 4c+o+1] = a[m, c>>1] * b[m, (c&1)*4 + o + 1]
        #pragma unroll
        for (int c = 0; c < 16; ++c) {
            const float ai = areg[c >> 1];
            v2f afrag;
            afrag.x = ai * ((c & 1) ? bl1.x : bl0.x);
            afrag.y = ai * ((c & 1) ? bl1.y : bl0.y);
            acc = __builtin_amdgcn_wmma_f32_16x16x4_f32(
                /*neg_a=*/false, afrag, /*neg_b=*/false, bfrag[c],
                /*c_mod=*/(short)0, acc, /*reuse_a=*/false, /*reuse_b=*/false);
        }

        // D layout: vgpr r -> row (r + half*8), column n. Full tile: store
        // unconditionally under the n<8 mask; lanes 0-7 / 16-23 each form a
        // contiguous 32B segment per store instruction.
        if (n < 8) {
            float* op = out + (t * 16 + half * 8) * 8 + n;
            #pragma unroll
            for (int r = 0; r < 8; ++r)
                op[r * 8] = acc[r];
        }
    }
}

// Scalar tail for n_mv % 16 leftover multivectors (not used by the harness
// batch size; kept for generality).
__global__ void clifford_gp_tail(
    const float* __restrict__ a,
    const float* __restrict__ b,
    const float* __restrict__ cayley,
    float* __restrict__ out,
    long long start, long long n_mv)
{
    const long long m = start + blockIdx.x * blockDim.x + threadIdx.x;
    if (m >= n_mv) return;
    float av[8], bv[8], ov[8];
    #pragma unroll
    for (int i = 0; i < 8; ++i) { av[i] = a[m * 8 + i]; bv[i] = b[m * 8 + i]; ov[i] = 0.0f; }
    for (int i = 0; i < 8; ++i)
        for (int j = 0; j < 8; ++j) {
            const float p = av[i] * bv[j];
            #pragma unroll
            for (int k = 0; k < 8; ++k)
                ov[k] = fmaf(p, cayley[i * 64 + j * 8 + k], ov[k]);
        }
    #pragma unroll
    for (int k = 0; k < 8; ++k) out[m * 8 + k] = ov[k];
}

extern "C" void kernel_launch(void* const* d_in, const int* in_sizes, int n_in,
                              void* d_out, int out_size, void* d_ws, size_t ws_size,
                              hipStream_t stream)
{
    const float* a      = (const float*)d_in[0];
    const float* b      = (const float*)d_in[1];
    const float* cayley = (const float*)d_in[2];
    float* out          = (float*)d_out;

    const long long n_mv    = (long long)in_sizes[0] / 8;   // 4,194,304
    const long long n_full  = n_mv / 16;                    // 262,144 full tiles
    const long long rem     = n_mv - n_full * 16;           // 0 for harness

    clifford_gp_wmma<<<4096, 256, 0, stream>>>(a, b, cayley, out, n_full);

    if (rem > 0)   // size-dependent only -> deterministic across calls
        clifford_gp_tail<<<1, 32, 0, stream>>>(a, b, cayley, out,
                                               n_full * 16, n_mv);
}